// Loss_Antonymy_34213709479938
// MI455X (gfx1250) — compile-verified
//
#include <hip/hip_runtime.h>

// MI455X (gfx1250, wave32). Bandwidth-bound streaming reduction (~516 MB @ 23.3 TB/s).
// Row sum-of-squares computed via V_WMMA_F32_16X16X4_F32 with an all-ones B matrix
// (exact f32 math; sum-of-squares is K-permutation invariant so lanes load
// contiguous float4s). Deterministic two-pass reduction via d_ws.

typedef float v2f __attribute__((ext_vector_type(2)));
typedef float v4f __attribute__((ext_vector_type(4)));
typedef float v8f __attribute__((ext_vector_type(8)));

#define DCOLS 64
#define TILE  16

__global__ void loss_antonymy_main(const float* __restrict__ S2,
                                   const float* __restrict__ A1,
                                   const int*  __restrict__ labels,
                                   float* __restrict__ partials,
                                   int nTiles, int nRows) {
    const int lane         = threadIdx.x & 31;
    const int waveInBlock  = threadIdx.x >> 5;
    const int wavesPerBlk  = blockDim.x >> 5;
    const int globalWave   = blockIdx.x * wavesPerBlk + waveInBlock;
    const int nWaves       = gridDim.x * wavesPerBlk;

    const int half = lane >> 4;   // 0: K=0,1 provider  1: K=2,3 provider
    const int lrow = lane & 15;   // row within 16-row tile that this lane feeds

    v2f ones; ones.x = 1.0f; ones.y = 1.0f;   // B matrix (4x16 of 1.0) = 2 VGPRs

    float part = 0.0f;

    for (int t = globalWave; t < nTiles; t += nWaves) {
        const int r0 = t * TILE;
        // Lane feeds row (r0+lrow); half 0 consumes elements 0..31, half 1: 32..63.
        const size_t base = (size_t)(r0 + lrow) * DCOLS + (size_t)half * 32;

        v4f a[8], s[8];
#pragma unroll
        for (int j = 0; j < 8; ++j) {
            a[j] = *(const v4f*)(A1 + base + 4 * j);   // global_load_b128
            s[j] = *(const v4f*)(S2 + base + 4 * j);   // global_load_b128
        }

        v8f c = {};   // 16x16 f32 accumulator (8 VGPRs)
#pragma unroll
        for (int j = 0; j < 8; ++j) {
            v4f dv = a[j] - s[j];
            v4f sq = dv * dv;
            v2f lo; lo.x = sq.x; lo.y = sq.y;
            v2f hi; hi.x = sq.z; hi.y = sq.w;
            // D[m][n] += sum_k A[m][k]  (B all ones) -> per-row partial sumsq
            c = __builtin_amdgcn_wmma_f32_16x16x4_f32(false, lo, false, ones,
                                                      (short)0, c, false, false);
            c = __builtin_amdgcn_wmma_f32_16x16x4_f32(false, hi, false, ones,
                                                      (short)0, c, false, false);
        }

        // C/D layout: lanes 0-15 VGPR r holds M=r, lanes 16-31 hold M=8+r.
        // Lane 0 -> rows r0..r0+7, lane 16 -> rows r0+8..r0+15. Others duplicate.
        if (lrow == 0) {
            const int rowBase = r0 + half * 8;
#pragma unroll
            for (int r = 0; r < 8; ++r) {
                float ss  = c[r];
                float d   = sqrtf(ss);
                float th  = tanhf(d);
                float sgn = (labels[rowBase + r] == 2) ? -1.0f : 1.0f;
                part += fmaxf(fmaf(sgn, th, 1.0f), 0.0f);
            }
        }
    }

    // Scalar tail for nRows % 16 (none for N=1048576, kept for safety).
    {
        const int gtid    = blockIdx.x * blockDim.x + threadIdx.x;
        const int threads = gridDim.x * blockDim.x;
        for (int row = nTiles * TILE + gtid; row < nRows; row += threads) {
            const size_t rb = (size_t)row * DCOLS;
            float ss = 0.0f;
            for (int k = 0; k < DCOLS; ++k) {
                float dv = A1[rb + k] - S2[rb + k];
                ss = fmaf(dv, dv, ss);
            }
            float d   = sqrtf(ss);
            float th  = tanhf(d);
            float sgn = (labels[row] == 2) ? -1.0f : 1.0f;
            part += fmaxf(fmaf(sgn, th, 1.0f), 0.0f);
        }
    }

    // Wave32 butterfly reduction, then per-block LDS reduction.
#pragma unroll
    for (int off = 16; off >= 1; off >>= 1)
        part += __shfl_xor(part, off, 32);

    __shared__ float sPart[32];
    if (lane == 0) sPart[waveInBlock] = part;
    __syncthreads();
    if (threadIdx.x == 0) {
        float b = 0.0f;
        for (int w = 0; w < wavesPerBlk; ++w) b += sPart[w];
        partials[blockIdx.x] = b;   // deterministic per-block partial
    }
}

__global__ void loss_antonymy_reduce(const float* __restrict__ partials, int n,
                                     float* __restrict__ out) {
    __shared__ float buf[256];
    float s = 0.0f;
    for (int i = threadIdx.x; i < n; i += blockDim.x) s += partials[i];
    buf[threadIdx.x] = s;
    __syncthreads();
    for (int stride = 128; stride > 0; stride >>= 1) {
        if ((int)threadIdx.x < stride) buf[threadIdx.x] += buf[threadIdx.x + stride];
        __syncthreads();
    }
    if (threadIdx.x == 0) out[0] = buf[0];   // overwrite (d_out is poisoned)
}

extern "C" void kernel_launch(void* const* d_in, const int* in_sizes, int n_in,
                              void* d_out, int out_size, void* d_ws, size_t ws_size,
                              hipStream_t stream) {
    const float* S2     = (const float*)d_in[0];
    const float* A1     = (const float*)d_in[1];
    const int*   labels = (const int*)d_in[2];
    float*       out    = (float*)d_out;
    float*       parts  = (float*)d_ws;

    const int nRows  = in_sizes[2];          // N (labels count)
    const int nTiles = nRows / TILE;

    int blocks = 1024;                       // 8192 waves -> 8 tiles/wave at N=1M
    int maxBlocks = (int)(ws_size / sizeof(float));
    if (blocks > maxBlocks) blocks = maxBlocks;
    if (blocks < 1) blocks = 1;

    loss_antonymy_main<<<blocks, 256, 0, stream>>>(S2, A1, labels, parts,
                                                   nTiles, nRows);
    loss_antonymy_reduce<<<1, 256, 0, stream>>>(parts, blocks, out);
}